// NodeModel_44573170598878
// MI455X (gfx1250) — compile-verified
//
#include <hip/hip_runtime.h>

// FP32 WMMA fragment types (CDNA5, wave32):
//   A 16x4 f32 -> 2 VGPRs/lane, B 4x16 f32 -> 2 VGPRs/lane, C/D 16x16 f32 -> 8 VGPRs/lane
typedef float v2f __attribute__((ext_vector_type(2)));
typedef float v8f __attribute__((ext_vector_type(8)));

#define F_NODE   64
#define F_EDGE   32
#define F_GLOB   32
#define OUT_DIM  64
#define K1       (F_NODE + F_EDGE)            // 96
#define K2       (F_NODE + OUT_DIM + F_GLOB)  // 160
#define LDS_STRIDE1 100   // 96 padded: row stride 400 B (16B mult), banks conflict-free
#define LDS_STRIDE2 164   // 160 padded: row stride 656 B (16B mult), banks conflict-free
#define ROWS_PER_BLOCK 64

// ---- CDNA5 async global->LDS copy (16 bytes), tracked by ASYNCcnt ----------
#define AS3 __attribute__((address_space(3)))
__device__ __forceinline__ uint32_t lds_byte_off(const void* p) {
    return (uint32_t)(uintptr_t)(AS3 const void*)p;   // addrspacecast -> 32-bit LDS offset
}
__device__ __forceinline__ void async_ld16(const float* g, uint32_t lds_off) {
    asm volatile("global_load_async_to_lds_b128 %0, %1, off"
                 :: "v"(lds_off), "v"((unsigned long long)(uintptr_t)g)
                 : "memory");
}
__device__ __forceinline__ void wait_async0() {
    asm volatile("s_wait_asynccnt 0x0" ::: "memory");
}

__global__ void zero_f32_kernel(float* __restrict__ p, int n) {
    int i = blockIdx.x * blockDim.x + threadIdx.x;
    if (i < n) p[i] = 0.0f;
}

// ---------------------------------------------------------------------------
// Kernel 1: edge MLP + scatter-add.
//   messages = relu(concat(x[src], edge_attr) @ W1 + b1)   [E, 64]
//   agg[dest] += messages                                   (hw fp atomics, L2-resident)
// Block = 256 thr = 8 waves; 64 edges/block; wave = 2 x (16 edges x 16 outs) tiles.
// ---------------------------------------------------------------------------
__global__ __launch_bounds__(256) void edge_mlp_scatter_kernel(
    const float* __restrict__ x,          // [N, 64]
    const int*   __restrict__ edge_index, // [2, E] (dest row, src row)
    const float* __restrict__ edge_attr,  // [E, 32]
    const float* __restrict__ W1,         // [96, 64]
    const float* __restrict__ b1,         // [64]
    float*       __restrict__ agg,        // [N, 64]
    int E)
{
    __shared__ __align__(16) float tileA[ROWS_PER_BLOCK * LDS_STRIDE1];

    const int tid   = threadIdx.x;
    const int ebase = blockIdx.x * ROWS_PER_BLOCK;
    const int* dest_idx = edge_index;
    const int* src_idx  = edge_index + E;

    // Async gather: 4 threads per edge row, 24 x 16B chunks per row (96 floats).
    {
        const int row = tid >> 2;          // 0..63
        const int t   = tid & 3;
        int e = ebase + row;
        if (e >= E) e = E - 1;             // clamp (E % 64 == 0 normally)
        const int    s  = src_idx[e];
        const float* xr = x + (long long)s * F_NODE;
        const float* ar = edge_attr + (long long)e * F_EDGE;
        const uint32_t lb = lds_byte_off(&tileA[row * LDS_STRIDE1]);
#pragma unroll
        for (int i = 0; i < 6; ++i) {
            const int c = t + 4 * i;       // chunk 0..23
            const float* g = (c < 16) ? (xr + c * 4) : (ar + (c - 16) * 4);
            async_ld16(g, lb + (uint32_t)c * 16u);
        }
    }

    const int wave = tid >> 5;
    const int lane = tid & 31;
    const int hl   = lane >> 4;        // 0: lanes 0-15, 1: lanes 16-31
    const int tm0  = wave >> 2;        // first M-tile (0..1); second = tm0+2
    const int tn   = wave & 3;         // N-tile (0..3)
    const int ncol = lane & 15;
    const int jcol = tn * 16 + ncol;   // global output column

    // Preload B fragments (overlaps with async gather):
    // VGPR j of step s holds W1[4s + 2*hl + j][jcol].
    v2f bfrag[K1 / 4];
#pragma unroll
    for (int s = 0; s < K1 / 4; ++s) {
        const int k = 4 * s + 2 * hl;
        bfrag[s].x = W1[(k + 0) * OUT_DIM + jcol];
        bfrag[s].y = W1[(k + 1) * OUT_DIM + jcol];
    }

    const float bias = b1[jcol];
    v8f acc0 = { bias, bias, bias, bias, bias, bias, bias, bias };
    v8f acc1 = acc0;

    wait_async0();
    __syncthreads();

    // A fragment: lane holds A[M = lane&15][K = 4s + 2*hl + {0,1}].
    const float* arow0 = &tileA[((tm0 + 0) * 16 + ncol) * LDS_STRIDE1];
    const float* arow1 = &tileA[((tm0 + 2) * 16 + ncol) * LDS_STRIDE1];
#pragma unroll
    for (int s = 0; s < K1 / 4; ++s) {
        const int k = 4 * s + 2 * hl;
        const v2f a0 = *(const v2f*)(arow0 + k);
        const v2f a1 = *(const v2f*)(arow1 + k);
        acc0 = __builtin_amdgcn_wmma_f32_16x16x4_f32(
                   false, a0, false, bfrag[s], (short)0, acc0, false, false);
        acc1 = __builtin_amdgcn_wmma_f32_16x16x4_f32(
                   false, a1, false, bfrag[s], (short)0, acc1, false, false);
    }

    // ReLU + scatter-add. D layout: VGPR r, lanes 0-15 -> M=r, 16-31 -> M=r+8.
#pragma unroll
    for (int r = 0; r < 8; ++r) {
        const int m = r + 8 * hl;
        const int e0 = ebase + (tm0 + 0) * 16 + m;
        if (e0 < E) {
            float v = acc0[r];
            v = v > 0.0f ? v : 0.0f;
            unsafeAtomicAdd(&agg[(long long)dest_idx[e0] * OUT_DIM + jcol], v);
        }
        const int e1 = ebase + (tm0 + 2) * 16 + m;
        if (e1 < E) {
            float v = acc1[r];
            v = v > 0.0f ? v : 0.0f;
            unsafeAtomicAdd(&agg[(long long)dest_idx[e1] * OUT_DIM + jcol], v);
        }
    }
}

// ---------------------------------------------------------------------------
// Kernel 2: node MLP.
//   out = relu(concat(x, agg, u[batch]) @ W2 + b2)   [N, 64]
// ---------------------------------------------------------------------------
__global__ __launch_bounds__(256) void node_mlp_kernel(
    const float* __restrict__ x,      // [N, 64]
    const float* __restrict__ agg,    // [N, 64]
    const float* __restrict__ u,      // [G, 32]
    const int*   __restrict__ batch,  // [N]
    const float* __restrict__ W2,     // [160, 64]
    const float* __restrict__ b2,     // [64]
    float*       __restrict__ out,    // [N, 64]
    int N)
{
    __shared__ __align__(16) float tileA[ROWS_PER_BLOCK * LDS_STRIDE2];

    const int tid   = threadIdx.x;
    const int nbase = blockIdx.x * ROWS_PER_BLOCK;

    // Async gather: 4 threads per node row, 40 x 16B chunks per row (160 floats).
    {
        const int row = tid >> 2;
        const int t   = tid & 3;
        int n = nbase + row;
        if (n >= N) n = N - 1;             // clamp; stores guarded below
        const int    g  = batch[n];
        const float* xr = x   + (long long)n * F_NODE;
        const float* gr = agg + (long long)n * OUT_DIM;
        const float* ur = u   + (long long)g * F_GLOB;
        const uint32_t lb = lds_byte_off(&tileA[row * LDS_STRIDE2]);
#pragma unroll
        for (int i = 0; i < 10; ++i) {
            const int c = t + 4 * i;       // chunk 0..39
            const float* gp;
            if (c < 16)      gp = xr + c * 4;
            else if (c < 32) gp = gr + (c - 16) * 4;
            else             gp = ur + (c - 32) * 4;
            async_ld16(gp, lb + (uint32_t)c * 16u);
        }
    }

    const int wave = tid >> 5;
    const int lane = tid & 31;
    const int hl   = lane >> 4;
    const int tm0  = wave >> 2;
    const int tn   = wave & 3;
    const int ncol = lane & 15;
    const int jcol = tn * 16 + ncol;

    v2f bfrag[K2 / 4];
#pragma unroll
    for (int s = 0; s < K2 / 4; ++s) {
        const int k = 4 * s + 2 * hl;
        bfrag[s].x = W2[(k + 0) * OUT_DIM + jcol];
        bfrag[s].y = W2[(k + 1) * OUT_DIM + jcol];
    }

    const float bias = b2[jcol];
    v8f acc0 = { bias, bias, bias, bias, bias, bias, bias, bias };
    v8f acc1 = acc0;

    wait_async0();
    __syncthreads();

    const float* arow0 = &tileA[((tm0 + 0) * 16 + ncol) * LDS_STRIDE2];
    const float* arow1 = &tileA[((tm0 + 2) * 16 + ncol) * LDS_STRIDE2];
#pragma unroll
    for (int s = 0; s < K2 / 4; ++s) {
        const int k = 4 * s + 2 * hl;
        const v2f a0 = *(const v2f*)(arow0 + k);
        const v2f a1 = *(const v2f*)(arow1 + k);
        acc0 = __builtin_amdgcn_wmma_f32_16x16x4_f32(
                   false, a0, false, bfrag[s], (short)0, acc0, false, false);
        acc1 = __builtin_amdgcn_wmma_f32_16x16x4_f32(
                   false, a1, false, bfrag[s], (short)0, acc1, false, false);
    }

#pragma unroll
    for (int r = 0; r < 8; ++r) {
        const int m = r + 8 * hl;
        const int n0 = nbase + (tm0 + 0) * 16 + m;
        if (n0 < N) {
            float v = acc0[r];
            out[(long long)n0 * OUT_DIM + jcol] = v > 0.0f ? v : 0.0f;
        }
        const int n1 = nbase + (tm0 + 2) * 16 + m;
        if (n1 < N) {
            float v = acc1[r];
            out[(long long)n1 * OUT_DIM + jcol] = v > 0.0f ? v : 0.0f;
        }
    }
}

// ---------------------------------------------------------------------------
extern "C" void kernel_launch(void* const* d_in, const int* in_sizes, int n_in,
                              void* d_out, int out_size, void* d_ws, size_t ws_size,
                              hipStream_t stream) {
    const float* x          = (const float*)d_in[0];
    const int*   edge_index = (const int*)  d_in[1];
    const float* edge_attr  = (const float*)d_in[2];
    const float* u          = (const float*)d_in[3];
    const int*   batch      = (const int*)  d_in[4];
    const float* W1         = (const float*)d_in[5];
    const float* b1         = (const float*)d_in[6];
    const float* W2         = (const float*)d_in[7];
    const float* b2         = (const float*)d_in[8];
    float* out = (float*)d_out;

    const int N = in_sizes[0] / F_NODE;   // 50000
    const int E = in_sizes[1] / 2;        // 800000

    float* agg = (float*)d_ws;            // [N, 64] f32 scratch (12.8 MB, L2-resident)
    const int nAgg = N * OUT_DIM;

    hipLaunchKernelGGL(zero_f32_kernel,
                       dim3((nAgg + 255) / 256), dim3(256), 0, stream, agg, nAgg);

    hipLaunchKernelGGL(edge_mlp_scatter_kernel,
                       dim3((E + ROWS_PER_BLOCK - 1) / ROWS_PER_BLOCK), dim3(256), 0, stream,
                       x, edge_index, edge_attr, W1, b1, agg, E);

    hipLaunchKernelGGL(node_mlp_kernel,
                       dim3((N + ROWS_PER_BLOCK - 1) / ROWS_PER_BLOCK), dim3(256), 0, stream,
                       x, agg, u, batch, W2, b2, out, N);
}